// GlobalMixer_47579647705676
// MI455X (gfx1250) — compile-verified
//
#include <hip/hip_runtime.h>
#include <hip/hip_bf16.h>

typedef __attribute__((ext_vector_type(2))) float v2f;
typedef __attribute__((ext_vector_type(8))) float v8f;

#define HIDDEN_C 512
#define BLOCK_C 16
#define GROUPS_C 16
#define SEQ_C 256
#define BATCH_C 512
#define TILES_PER_WAVE 8
#define WAVES_PER_BLOCK 8

// y[b,h,o,g] = sum_i W[h,o,i] * x[b,h,i,g]
// One wave handles 8 consecutive batch tiles for a fixed h, reusing the
// W[h] A-fragments in registers. Each 16x16x16 product = 4 chained
// V_WMMA_F32_16X16X4_F32 ops (fp32 in, fp32 accumulate — exact-precision path;
// kernel is HBM-bound at ~4 FLOP/byte so fp32 WMMA costs nothing).
__global__ __launch_bounds__(256) void GlobalMixer_47579647705676_kernel(
    const float* __restrict__ x, const float* __restrict__ W,
    float* __restrict__ y) {
  const int lane = threadIdx.x & 31;
  const int waveInBlock = threadIdx.x >> 5;
  const int wave = blockIdx.x * WAVES_PER_BLOCK + waveInBlock;

  const int h = wave % HIDDEN_C;
  const int b0 = (wave / HIDDEN_C) * TILES_PER_WAVE;

  const int col = lane & 15;        // N (and M for A rows)
  const int hiHalf = lane >> 4;     // 0: lanes 0-15, 1: lanes 16-31

  // ---- A fragments: W[h] as four 16x4 slices -------------------------------
  // 32-bit A 16x4 layout: lanes 0-15 hold (K=0,K=1), lanes 16-31 hold (K=2,K=3)
  // => lane reads W[h][col][4*kb + 2*hiHalf .. +1], contiguous 8 bytes.
  const float* __restrict__ Wh = W + (size_t)h * (BLOCK_C * BLOCK_C);
  v2f a[4];
#pragma unroll
  for (int kb = 0; kb < 4; ++kb) {
    const float* p = Wh + col * BLOCK_C + 4 * kb + 2 * hiHalf;
    v2f av;
    av.x = p[0];
    av.y = p[1];
    a[kb] = av;
  }

#pragma unroll 1
  for (int bi = 0; bi < TILES_PER_WAVE; ++bi) {
    const int b = b0 + bi;
    const size_t tileOff = ((size_t)b * HIDDEN_C + h) * SEQ_C;
    const float* __restrict__ xt = x + tileOff;
    float* __restrict__ yt = y + tileOff;

    // ---- B fragments: X rows as four 4x16 slices ---------------------------
    // 32-bit B 4x16 layout (mirror of A): lane holds column n=col;
    // VGPR0: K = 4kb + 2*hiHalf, VGPR1: K = 4kb + 2*hiHalf + 1.
    // Streamed once -> non-temporal loads (don't pollute 192MB L2).
    v2f bfrag[4];
#pragma unroll
    for (int kb = 0; kb < 4; ++kb) {
      const int r = 4 * kb + 2 * hiHalf;
      v2f bv;
      bv.x = __builtin_nontemporal_load(xt + r * GROUPS_C + col);
      bv.y = __builtin_nontemporal_load(xt + (r + 1) * GROUPS_C + col);
      bfrag[kb] = bv;
    }

    // ---- 16x16x16 matmul as 4 chained K=4 WMMAs ----------------------------
    v8f c = {};
#pragma unroll
    for (int kb = 0; kb < 4; ++kb) {
      c = __builtin_amdgcn_wmma_f32_16x16x4_f32(
          /*neg_a=*/false, a[kb], /*neg_b=*/false, bfrag[kb],
          /*c_mod=*/(short)0, c, /*reuse_a=*/false, /*reuse_b=*/false);
    }

    // ---- Store D: VGPR j -> Y[j + 8*hiHalf][col], non-temporal stream ------
#pragma unroll
    for (int j = 0; j < 8; ++j) {
      __builtin_nontemporal_store(c[j], yt + (j + 8 * hiHalf) * GROUPS_C + col);
    }
  }
}

extern "C" void kernel_launch(void* const* d_in, const int* in_sizes, int n_in,
                              void* d_out, int out_size, void* d_ws, size_t ws_size,
                              hipStream_t stream) {
  (void)in_sizes; (void)n_in; (void)d_ws; (void)ws_size; (void)out_size;
  const float* x = (const float*)d_in[0];  // (512, 512, 1, 256) fp32
  const float* W = (const float*)d_in[1];  // (512, 16, 16) fp32
  float* y = (float*)d_out;                // (512, 512, 1, 256) fp32

  const int totalWaves = (BATCH_C * HIDDEN_C) / TILES_PER_WAVE;  // 32768
  const int blocks = totalWaves / WAVES_PER_BLOCK;               // 4096
  GlobalMixer_47579647705676_kernel<<<blocks, WAVES_PER_BLOCK * 32, 0, stream>>>(x, W, y);
}